// digitCapsule_63075889709358
// MI455X (gfx1250) — compile-verified
//
#include <hip/hip_runtime.h>
#include <math.h>

// CDNA5 wave32 WMMA fragment types
typedef __attribute__((ext_vector_type(2))) float v2f;
typedef __attribute__((ext_vector_type(4))) float v4f;
typedef __attribute__((ext_vector_type(8))) float v8f;

#define B_SZ    128
#define D_IN    8
#define N_OUT   10
#define D_OUT   16
#define NPOS    4608          // 32 capsules * 12 * 12
#define NCOL    160           // D_OUT * N_OUT  (flattened o*10+j)
#define NUM_ROUTING 3
#define NCHUNK  32            // n positions per block
#define NCHUNKS (NPOS / NCHUNK)   // 144
#define BG      32            // batches per block (2 wave32 b-tiles)
#define BGROUPS (B_SZ / BG)       // 4

// fragment element -> K/D index: c=0 -> 2*kh, c=1 -> 2*kh+1, c=2 -> 2*kh+4, c=3 -> 2*kh+5
__device__ __forceinline__ int frag_k(int khalf, int c) {
    return 2 * khalf + (c & 1) + (c >> 1) * 4;
}

// ---------------------------------------------------------------------------
// Pack x (B, D_IN, NPOS) into per-lane A-fragment order:
//   upk[((n*B + b)*2 + khalf)*4 + c] = u[b,n,d] with d = frag_k(khalf,c)
// One b128 load then yields both 16x4 A-fragments for a lane.
// ---------------------------------------------------------------------------
__global__ void caps_pack_u(const float* __restrict__ x, float* __restrict__ upk) {
    int tid = blockIdx.x * blockDim.x + threadIdx.x;
    const int total = NPOS * B_SZ * D_IN;
    if (tid >= total) return;
    const int c     = tid & 3;
    const int khalf = (tid >> 2) & 1;
    const int b     = (tid >> 3) % B_SZ;
    const int n     = tid / (B_SZ * D_IN);
    const int d     = frag_k(khalf, c);
    upk[tid] = x[((size_t)b * D_IN + d) * NPOS + n];
}

// ---------------------------------------------------------------------------
// Pack weight (D_IN, D_OUT, NPOS, N_OUT) into per-lane B-fragment order:
//   wpk[(((n*10 + t)*32 + lane)*4 + c] = w[k, o, n, j]
//   with m=lane&15, khalf=lane>>4, col=t*16+m, o=col/10, j=col%10, k=frag_k(khalf,c)
// One b128 load then yields both 4x16 B-fragments of tile t for a lane.
// ---------------------------------------------------------------------------
__global__ void caps_pack_w(const float* __restrict__ w, float* __restrict__ wpk) {
    int tid = blockIdx.x * blockDim.x + threadIdx.x;
    const int total = NPOS * 10 * 32 * 4;
    if (tid >= total) return;
    const int c     = tid & 3;
    const int lane  = (tid >> 2) & 31;
    const int t     = (tid >> 7) % 10;
    const int n     = tid / (10 * 128);
    const int m     = lane & 15;
    const int khalf = lane >> 4;
    const int col   = t * 16 + m;
    const int oo    = col / N_OUT;
    const int jj    = col - oo * N_OUT;
    const int k     = frag_k(khalf, c);
    wpk[tid] = w[((size_t)(k * D_OUT + oo) * NPOS + n) * N_OUT + jj];
}

// ---------------------------------------------------------------------------
// Fused routing pass: recompute u_hat tiles with V_WMMA_F32_16X16X4_F32 from
// packed fragments (all loads coalesced b128), form logits from running S_acc,
// softmax over j, accumulate partial s.
// grid = (NCHUNKS, BGROUPS), block = 64 (2 waves; wave w owns one 16-batch tile)
// ---------------------------------------------------------------------------
__global__ void __launch_bounds__(64)
caps_route(const float* __restrict__ upk, const float* __restrict__ wpk,
           const float* __restrict__ Sacc, float* __restrict__ spart, int iter)
{
    __shared__ float uhat_s[BG][NCOL];   // 20 KB  u_hat tile for 32 batches
    __shared__ float sacc_s[BG][NCOL];   // 20 KB  S_acc slice (logit weights)
    __shared__ float spart_s[BG][NCOL];  // 20 KB  per-block partial s accumulator

    const int tid   = threadIdx.x;
    const int wave  = tid >> 5;
    const int lane  = tid & 31;
    const int b0    = blockIdx.y * BG;       // global batch base of this block
    const int n0    = blockIdx.x * NCHUNK;

    for (int i = tid; i < BG * NCOL; i += 64) {
        (&spart_s[0][0])[i] = 0.0f;
        (&sacc_s[0][0])[i]  = (iter > 0) ? Sacc[(size_t)b0 * NCOL + i] : 0.0f;
    }
    __syncthreads();

    const int m     = lane & 15;   // M index (A) / N index (B) within tile
    const int khalf = lane >> 4;   // lane half selects K pairs
    const int bt    = wave * 16;   // local b-tile base

    for (int nn = 0; nn < NCHUNK; ++nn) {
        const int n = n0 + nn;

        // A fragments: one coalesced b128 per lane
        const v4f av = *(const v4f*)(upk +
            (((size_t)n * B_SZ + (b0 + bt + m)) * 2 + khalf) * 4);
        const v2f a0 = { av.x, av.y };   // K 2kh,2kh+1
        const v2f a1 = { av.z, av.w };   // K 2kh+4,2kh+5

        const float* wn = wpk + ((size_t)n * 10) * 128;  // this n's packed tiles

#pragma unroll
        for (int t = 0; t < 10; ++t) {
            // B fragments: one coalesced b128 per lane per tile
            const v4f bv = *(const v4f*)(wn + ((size_t)t * 32 + lane) * 4);
            const v2f bf0 = { bv.x, bv.y };
            const v2f bf1 = { bv.z, bv.w };

            v8f acc = {};
            acc = __builtin_amdgcn_wmma_f32_16x16x4_f32(false, a0, false, bf0,
                                                        (short)0, acc, false, false);
            acc = __builtin_amdgcn_wmma_f32_16x16x4_f32(false, a1, false, bf1,
                                                        (short)0, acc, false, false);
            // C/D layout: VGPR r -> M = r (lanes 0-15) / r+8 (lanes 16-31), N = m
            const int col = t * 16 + m;
#pragma unroll
            for (int r = 0; r < 8; ++r)
                uhat_s[bt + r + khalf * 8][col] = acc[r];
        }
        __syncthreads();

        // Routing math: lanes of wave 0 each own one batch row of this block
        if (tid < BG) {
            const int bl = tid;
            float cj[N_OUT];
            if (iter == 0) {
#pragma unroll
                for (int j = 0; j < N_OUT; ++j) cj[j] = 0.1f;  // softmax(0)
            } else {
                float lg[N_OUT];
#pragma unroll
                for (int j = 0; j < N_OUT; ++j) lg[j] = 0.0f;
#pragma unroll
                for (int o = 0; o < D_OUT; ++o)
#pragma unroll
                    for (int j = 0; j < N_OUT; ++j)
                        lg[j] += uhat_s[bl][o * N_OUT + j] * sacc_s[bl][o * N_OUT + j];
                float mx = lg[0];
#pragma unroll
                for (int j = 1; j < N_OUT; ++j) mx = fmaxf(mx, lg[j]);
                float sum = 0.0f;
#pragma unroll
                for (int j = 0; j < N_OUT; ++j) { cj[j] = __expf(lg[j] - mx); sum += cj[j]; }
                const float inv = 1.0f / sum;
#pragma unroll
                for (int j = 0; j < N_OUT; ++j) cj[j] *= inv;
            }
#pragma unroll
            for (int o = 0; o < D_OUT; ++o)
#pragma unroll
                for (int j = 0; j < N_OUT; ++j)
                    spart_s[bl][o * N_OUT + j] += cj[j] * uhat_s[bl][o * N_OUT + j];
        }
        __syncthreads();
    }

    // write this block's partial slab: spart[chunk][b][col] (fully owned -> no pre-zero)
    float* slab = spart + ((size_t)blockIdx.x * B_SZ + b0) * NCOL;
    for (int i = tid; i < BG * NCOL; i += 64)
        slab[i] = (&spart_s[0][0])[i];
}

// ---------------------------------------------------------------------------
// Reduce partials over chunks, squash over D_OUT, update S_acc, emit output.
// One thread per (b, j); fixed-order reduction -> deterministic.
// ---------------------------------------------------------------------------
__global__ void caps_squash(const float* __restrict__ spart, float* __restrict__ Sacc,
                            float* __restrict__ out, int iter)
{
    int tid = blockIdx.x * blockDim.x + threadIdx.x;
    if (tid >= B_SZ * N_OUT) return;
    const int b = tid / N_OUT;
    const int j = tid - b * N_OUT;

    float s[D_OUT];
#pragma unroll
    for (int o = 0; o < D_OUT; ++o) s[o] = 0.0f;
    for (int ch = 0; ch < NCHUNKS; ++ch) {
        const float* p = spart + ((size_t)ch * B_SZ + b) * NCOL + j;
#pragma unroll
        for (int o = 0; o < D_OUT; ++o) s[o] += p[o * N_OUT];
    }
    float ssq = 0.0f;
#pragma unroll
    for (int o = 0; o < D_OUT; ++o) ssq += s[o] * s[o];
    const float f = ssq / ((1.0f + ssq) * sqrtf(ssq));  // squash factor (no eps, as ref)
#pragma unroll
    for (int o = 0; o < D_OUT; ++o) {
        const float v = f * s[o];
        const size_t idx = ((size_t)b * D_OUT + o) * N_OUT + j;
        if (iter == 0) Sacc[idx] = v; else Sacc[idx] += v;   // running logit weights
        if (iter == NUM_ROUTING - 1) out[idx] = v;           // (B, D_OUT, N_OUT)
    }
}

// ---------------------------------------------------------------------------
extern "C" void kernel_launch(void* const* d_in, const int* in_sizes, int n_in,
                              void* d_out, int out_size, void* d_ws, size_t ws_size,
                              hipStream_t stream)
{
    const float* x = (const float*)d_in[0];   // (128, 8, 32, 12, 12)
    const float* w = (const float*)d_in[1];   // (8, 16, 4608, 10)
    float* out = (float*)d_out;               // (128, 16, 10)

    // workspace layout (~54.4 MB total)
    char*  ws    = (char*)d_ws;
    float* upk   = (float*)ws;                                            // 18.9 MB
    float* wpk   = (float*)(ws + (size_t)NPOS * B_SZ * D_IN * 4);         // 23.6 MB
    float* spart = (float*)((char*)wpk + (size_t)NPOS * 10 * 128 * 4);    // 11.8 MB
    float* Sacc  = (float*)((char*)spart + (size_t)NCHUNKS * B_SZ * NCOL * 4); // 80 KB

    {
        const int totu = NPOS * B_SZ * D_IN;
        caps_pack_u<<<(totu + 255) / 256, 256, 0, stream>>>(x, upk);
        const int totw = NPOS * 10 * 32 * 4;
        caps_pack_w<<<(totw + 255) / 256, 256, 0, stream>>>(w, wpk);
    }
    for (int iter = 0; iter < NUM_ROUTING; ++iter) {
        caps_route<<<dim3(NCHUNKS, BGROUPS), 64, 0, stream>>>(upk, wpk, Sacc, spart, iter);
        caps_squash<<<(B_SZ * N_OUT + 255) / 256, 256, 0, stream>>>(spart, Sacc, out, iter);
    }
}